// GroupListNetLoss_24275155157726
// MI455X (gfx1250) — compile-verified
//
#include <hip/hip_runtime.h>

// ---------------------------------------------------------------------------
// GroupListNetLoss on MI455X (gfx1250)
//
// loss_g = log(sum exp(p - pmax)) + pmax - (sum exp(t - tmax) * p) / (sum exp(t - tmax))
// output = mean over groups with count >= 2.
//
// Memory-bound: 48 MB inputs, read twice; second pass is L2-resident (192 MB L2).
// Per-group tables (128 KB each) accumulate via L2 atomics.
// Final 32K-element reduction uses V_WMMA_F32_16X16X4_F32 (fp32-exact, one wave).
// ---------------------------------------------------------------------------

#define NGROUPS 32768

typedef float v2f __attribute__((ext_vector_type(2)));
typedef float v8f __attribute__((ext_vector_type(8)));

// Monotonic float <-> uint mapping so float-max can use native integer atomicMax.
// Zero is the identity element (maps back only to -NaN, never produced by inputs).
__device__ __forceinline__ unsigned ordf(float f) {
    unsigned u = __float_as_uint(f);
    return (u & 0x80000000u) ? ~u : (u | 0x80000000u);
}
__device__ __forceinline__ float unordf(unsigned v) {
    unsigned u = (v & 0x80000000u) ? (v & 0x7fffffffu) : ~v;
    return __uint_as_float(u);
}

// ---- Pass 1: per-group max(t), max(p), count --------------------------------
__global__ __launch_bounds__(256) void k_pass1(const float4* __restrict__ pred4,
                                               const float4* __restrict__ targ4,
                                               const int4*   __restrict__ gid4,
                                               unsigned* __restrict__ tmaxo,
                                               unsigned* __restrict__ pmaxo,
                                               unsigned* __restrict__ cnt,
                                               int n4) {
    int i = blockIdx.x * blockDim.x + threadIdx.x;
    if (i >= n4) return;
    float4 p = pred4[i];
    float4 t = targ4[i];
    int4   g = gid4[i];
    atomicMax(&tmaxo[g.x], ordf(t.x)); atomicMax(&pmaxo[g.x], ordf(p.x)); atomicAdd(&cnt[g.x], 1u);
    atomicMax(&tmaxo[g.y], ordf(t.y)); atomicMax(&pmaxo[g.y], ordf(p.y)); atomicAdd(&cnt[g.y], 1u);
    atomicMax(&tmaxo[g.z], ordf(t.z)); atomicMax(&pmaxo[g.z], ordf(p.z)); atomicAdd(&cnt[g.z], 1u);
    atomicMax(&tmaxo[g.w], ordf(t.w)); atomicMax(&pmaxo[g.w], ordf(p.w)); atomicAdd(&cnt[g.w], 1u);
}

// ---- Pass 2: per-group sum exp(t-tmax), sum exp(p-pmax), sum exp(t-tmax)*p --
__global__ __launch_bounds__(256) void k_pass2(const float4* __restrict__ pred4,
                                               const float4* __restrict__ targ4,
                                               const int4*   __restrict__ gid4,
                                               const unsigned* __restrict__ tmaxo,
                                               const unsigned* __restrict__ pmaxo,
                                               float* __restrict__ tsum,
                                               float* __restrict__ psum,
                                               float* __restrict__ stp,
                                               int n4) {
    int i = blockIdx.x * blockDim.x + threadIdx.x;
    if (i >= n4) return;
    float4 p = pred4[i];
    float4 t = targ4[i];
    int4   g = gid4[i];

    #pragma unroll
    for (int c = 0; c < 4; ++c) {
        int   gi = (c == 0) ? g.x : (c == 1) ? g.y : (c == 2) ? g.z : g.w;
        float pc = (c == 0) ? p.x : (c == 1) ? p.y : (c == 2) ? p.z : p.w;
        float tc = (c == 0) ? t.x : (c == 1) ? t.y : (c == 2) ? t.z : t.w;
        float tm = unordf(tmaxo[gi]);
        float pm = unordf(pmaxo[gi]);
        float te = __expf(tc - tm);   // <= 1, stable
        float pe = __expf(pc - pm);
        atomicAdd(&tsum[gi], te);
        atomicAdd(&psum[gi], pe);
        atomicAdd(&stp[gi],  te * pc);
    }
}

// ---- Pass 3: per-group loss (0 if count < 2), count valid groups ------------
__global__ __launch_bounds__(256) void k_pass3(const unsigned* __restrict__ cnt,
                                               const unsigned* __restrict__ pmaxo,
                                               const float* __restrict__ tsum,
                                               const float* __restrict__ psum,
                                               const float* __restrict__ stp,
                                               float* __restrict__ losses,
                                               unsigned* __restrict__ nvalid) {
    int g = blockIdx.x * blockDim.x + threadIdx.x;
    if (g >= NGROUPS) return;
    unsigned c = cnt[g];
    float L = 0.0f;
    if (c >= 2u) {
        float pm = unordf(pmaxo[g]);
        // p_lse - sum(target_prob * p)
        L = logf(psum[g]) + pm - stp[g] / tsum[g];
        atomicAdd(nvalid, 1u);   // integer atomic: deterministic
    }
    losses[g] = L;
}

// ---- Pass 4: single-wave WMMA reduction of the 32768 losses -----------------
// A = 16x4 tile of losses (2 VGPRs/lane, fp32), B = all-ones 4x16.
// D[m][n] = sum_k A[m][k]; accumulate in C across 512 iterations, all fp32.
// C/D layout: lane 0 column N=0 holds M=0..7 in c[0..7]; lane 16 holds M=8..15.
__global__ __launch_bounds__(32) void k_reduce_wmma(const float* __restrict__ losses,
                                                    const unsigned* __restrict__ nvalid,
                                                    float* __restrict__ out) {
    const int lane = threadIdx.x;            // 0..31, EXEC all ones (WMMA requirement)
    v8f c = {0.f, 0.f, 0.f, 0.f, 0.f, 0.f, 0.f, 0.f};
    v2f bones = {1.0f, 1.0f};

    for (int base = 0; base < NGROUPS; base += 64) {
        v2f a = *(const v2f*)(losses + base + 2 * lane);   // 64 values per WMMA
        c = __builtin_amdgcn_wmma_f32_16x16x4_f32(
                /*neg_a=*/false, a, /*neg_b=*/false, bones,
                /*c_mod=*/(short)0, c, /*reuse_a=*/false, /*reuse_b=*/false);
    }

    // Column N=0: lane 0 has M=0..7, lane 16 has M=8..15.
    float s = c[0] + c[1] + c[2] + c[3] + c[4] + c[5] + c[6] + c[7];
    float total = __shfl(s, 0, 32) + __shfl(s, 16, 32);

    if (lane == 0) {
        unsigned nv = *nvalid;
        out[0] = (nv > 0u) ? (total / (float)nv) : 0.0f;
    }
}

// ---------------------------------------------------------------------------
extern "C" void kernel_launch(void* const* d_in, const int* in_sizes, int n_in,
                              void* d_out, int out_size, void* d_ws, size_t ws_size,
                              hipStream_t stream) {
    const float* pred = (const float*)d_in[0];
    const float* targ = (const float*)d_in[1];
    const int*   gid  = (const int*)d_in[2];
    const int n  = in_sizes[0];          // 4,194,304 (divisible by 4)
    const int n4 = n / 4;

    const unsigned G = NGROUPS;
    unsigned* tmaxo  = (unsigned*)d_ws;
    unsigned* pmaxo  = tmaxo + G;
    unsigned* cnt    = pmaxo + G;
    float*    tsum   = (float*)(cnt + G);
    float*    psum   = tsum + G;
    float*    stp    = psum + G;
    float*    losses = stp + G;
    unsigned* nvalid = (unsigned*)(losses + G);

    // Zero-init workspace: 0 is the identity for ordered-max, sums, counts.
    hipMemsetAsync(d_ws, 0, (size_t)(7 * G + 1) * sizeof(unsigned), stream);

    const int TB = 256;
    k_pass1<<<(n4 + TB - 1) / TB, TB, 0, stream>>>(
        (const float4*)pred, (const float4*)targ, (const int4*)gid,
        tmaxo, pmaxo, cnt, n4);

    k_pass2<<<(n4 + TB - 1) / TB, TB, 0, stream>>>(
        (const float4*)pred, (const float4*)targ, (const int4*)gid,
        tmaxo, pmaxo, tsum, psum, stp, n4);

    k_pass3<<<(G + TB - 1) / TB, TB, 0, stream>>>(
        cnt, pmaxo, tsum, psum, stp, losses, nvalid);

    k_reduce_wmma<<<1, 32, 0, stream>>>(losses, nvalid, (float*)d_out);
}